// SyntacticGCN_38774964748866
// MI455X (gfx1250) — compile-verified
//
#include <hip/hip_runtime.h>

typedef __attribute__((ext_vector_type(2))) float v2f;
typedef __attribute__((ext_vector_type(8))) float v8f;

namespace {
constexpr int kNodes         = 2048;  // B*S
constexpr int kNbr           = 128;   // M (max neighbors)
constexpr int kD             = 256;   // embedding dim
constexpr int kK             = 512;   // 2*D (GEMM K)
constexpr int kH             = 256;   // hidden dim (GEMM N)
constexpr int kNodesPerBlock = 16;    // GEMM M-tile
constexpr int kThreads       = 512;   // 16 wave32s
constexpr int kHPad          = 516;   // 512 + 4 pad -> conflict-free WMMA A reads
}

__global__ __launch_bounds__(512) void syngcn_fused_kernel(
    const float* __restrict__ src, const float* __restrict__ ngh,
    const float* __restrict__ Wm, const float* __restrict__ bm,
    float* __restrict__ out)
{
  __shared__ float    sh[kNodesPerBlock][kHPad];  // hidden tile [16][512] (+pad)
  __shared__ float4   ps[8][64];                  // src partials
  __shared__ float4   pn[8][64];                  // neigh partials
  __shared__ unsigned msk[4];                     // 128-bit row-validity mask

  const int tid  = threadIdx.x;
  const int lane = tid & 31;
  const int c4   = tid & 63;   // float4 column (0..63 -> cols 4c..4c+3)
  const int mg   = tid >> 6;   // neighbor-row group (0..7)

  const size_t node0 = (size_t)blockIdx.x * kNodesPerBlock;

  // ---------------- Phase 1: streaming reductions over neighbors ----------------
  for (int n = 0; n < kNodesPerBlock; ++n) {
    const size_t base = (node0 + (size_t)n) * (size_t)(kNbr * kD);
    const float4* sp = reinterpret_cast<const float4*>(src + base);
    const float4* np = reinterpret_cast<const float4*>(ngh + base);
    if (tid < 4) msk[tid] = 0u;
    __syncthreads();

    float4 sa = {0.f, 0.f, 0.f, 0.f};
    float4 na = {0.f, 0.f, 0.f, 0.f};
    #pragma unroll 4
    for (int m = mg; m < kNbr; m += 8) {
      const float4 sv = sp[m * (kD / 4) + c4];
      const float4 nv = np[m * (kD / 4) + c4];
      sa.x += sv.x; sa.y += sv.y; sa.z += sv.z; sa.w += sv.w;
      na.x += nv.x; na.y += nv.y; na.z += nv.z; na.w += nv.w;
      const bool nz = (nv.x != 0.f) || (nv.y != 0.f) || (nv.z != 0.f) || (nv.w != 0.f);
      const unsigned bal = (unsigned)__ballot(nz);   // wave32: low 32 bits
      if (lane == 0 && bal != 0u) atomicOr(&msk[m >> 5], 1u << (m & 31));
    }
    ps[mg][c4] = sa;
    pn[mg][c4] = na;
    __syncthreads();

    if (tid < 64) {
      float4 s  = ps[0][tid];
      float4 t4 = pn[0][tid];
      #pragma unroll
      for (int g = 1; g < 8; ++g) {
        const float4 a = ps[g][tid]; s.x += a.x;  s.y += a.y;  s.z += a.z;  s.w += a.w;
        const float4 b = pn[g][tid]; t4.x += b.x; t4.y += b.y; t4.z += b.z; t4.w += b.w;
      }
      const float num = (float)(__popc(msk[0]) + __popc(msk[1]) +
                                __popc(msk[2]) + __popc(msk[3]));
      const float inv = 1.0f / fmaxf(num, 1.0f);
      const int c = tid * 4;
      sh[n][c + 0] = s.x; sh[n][c + 1] = s.y; sh[n][c + 2] = s.z; sh[n][c + 3] = s.w;
      sh[n][kD + c + 0] = t4.x * inv;
      sh[n][kD + c + 1] = t4.y * inv;
      sh[n][kD + c + 2] = t4.z * inv;
      sh[n][kD + c + 3] = t4.w * inv;
    }
    __syncthreads();
  }

  // -------- Phase 2: out[16x256] = leaky_relu(sh[16x512] @ W[512x256] + b) --------
  // One wave per 16-wide N tile; V_WMMA_F32_16X16X4_F32, K stepped by 4.
  const int wv   = tid >> 5;        // wave id == N-tile id (0..15)
  const int half = lane >> 4;       // lane half selects K pair (A) / K row pair (B)
  const int l16  = lane & 15;
  const int col  = wv * 16 + l16;

  v8f c0 = {};
  v8f c1 = {};
  #pragma unroll 4
  for (int k = 0; k < kK; k += 8) {
    {
      const int kk = k + 2 * half;
      v2f a; a[0] = sh[l16][kk];                 a[1] = sh[l16][kk + 1];
      v2f b; b[0] = Wm[(size_t)kk * kH + col];   b[1] = Wm[(size_t)(kk + 1) * kH + col];
      c0 = __builtin_amdgcn_wmma_f32_16x16x4_f32(false, a, false, b,
                                                 (short)0, c0, false, false);
    }
    {
      const int kk = k + 4 + 2 * half;
      v2f a; a[0] = sh[l16][kk];                 a[1] = sh[l16][kk + 1];
      v2f b; b[0] = Wm[(size_t)kk * kH + col];   b[1] = Wm[(size_t)(kk + 1) * kH + col];
      c1 = __builtin_amdgcn_wmma_f32_16x16x4_f32(false, a, false, b,
                                                 (short)0, c1, false, false);
    }
  }

  const float bv = bm[col];
  #pragma unroll
  for (int r = 0; r < 8; ++r) {
    float v = c0[r] + c1[r] + bv;
    v = (v > 0.f) ? v : 0.01f * v;
    out[(node0 + (size_t)(half * 8 + r)) * kH + col] = v;
  }
}

extern "C" void kernel_launch(void* const* d_in, const int* in_sizes, int n_in,
                              void* d_out, int out_size, void* d_ws, size_t ws_size,
                              hipStream_t stream) {
  (void)in_sizes; (void)n_in; (void)d_ws; (void)ws_size; (void)out_size;
  const float* src = (const float*)d_in[0];   // [16,128,128,256] f32
  const float* ngh = (const float*)d_in[1];   // [16,128,128,256] f32
  const float* Wm  = (const float*)d_in[2];   // [512,256] f32
  const float* bm  = (const float*)d_in[3];   // [256] f32
  float* out = (float*)d_out;                 // [2048,256] f32

  dim3 grid(kNodes / kNodesPerBlock);         // 128 blocks
  dim3 block(kThreads);                       // 512 threads = 16 wave32s
  hipLaunchKernelGGL(syngcn_fused_kernel, grid, block, 0, stream,
                     src, ngh, Wm, bm, out);
}